// GATDiscriminator_31774168056072
// MI455X (gfx1250) — compile-verified
//
#include <hip/hip_runtime.h>
#include <hip/hip_bf16.h>
#include <stdint.h>

#define NNODES 100000
#define NEDGES 1600000
#define EPLUS  (NEDGES + NNODES)
#define NEG_SLOPE 0.2f
#define RR_SLOPE  0.2291666666666667f   /* (1/8 + 1/3)/2, torch.rrelu eval */

typedef float v2f __attribute__((ext_vector_type(2)));
typedef float v8f __attribute__((ext_vector_type(8)));

__device__ __forceinline__ float wave_sum32(float v) {
#pragma unroll
  for (int off = 16; off > 0; off >>= 1) v += __shfl_xor(v, off, 32);
  return v;
}

// ---------------------------------------------------------------------------
// GEMM: C[M x 192] = A[M x K] * B[K x 192], row-major, via V_WMMA_F32_16X16X4_F32.
// One wave per 16x16 output tile. SPLIT=true: A is the virtual concat of two
// width-64 row-major matrices A0 | A1 (K=128). SPLIT=false: A = A0, width K.
// Fragment layout per CDNA5 ISA 7.12.2 (wave32):
//   A 16x4 : lanes 0-15 -> M=lane, VGPR0=K0,VGPR1=K1 ; lanes 16-31 -> K2,K3
//   B 4x16 : lane%16 = N,  VGPR0/1 = the two K rows for this half-wave
//   C 16x16: VGPR j -> M = j + 8*(lane>=16), N = lane%16
// ---------------------------------------------------------------------------
template <int K, bool SPLIT>
__global__ void gemm_wmma_f32(const float* __restrict__ A0,
                              const float* __restrict__ A1,
                              const float* __restrict__ B,
                              float* __restrict__ C, int M) {
  const int lane  = threadIdx.x & 31;
  const int wave  = threadIdx.x >> 5;
  const int tile  = blockIdx.x * (blockDim.x >> 5) + wave;
  const int mtile = tile / 12;
  const int ntile = tile % 12;
  if (mtile >= (M >> 4)) return;

  const int row   = (mtile << 4) + (lane & 15);   // A row for this lane
  const int col   = (ntile << 4) + (lane & 15);   // B/C column for this lane
  const int khalf = (lane >> 4) << 1;             // 0 or 2

  v8f acc = {};
#pragma unroll 4
  for (int k = 0; k < K; k += 4) {
    const int ka = k + khalf;
    v2f a, b;
    if (SPLIT) {
      const float* ap = (ka < 64) ? (A0 + (size_t)row * 64 + ka)
                                  : (A1 + (size_t)row * 64 + (ka - 64));
      a[0] = ap[0]; a[1] = ap[1];
    } else {
      const float* ap = A0 + (size_t)row * K + ka;
      a[0] = ap[0]; a[1] = ap[1];
    }
    const float* bp = B + (size_t)ka * 192 + col;
    b[0] = bp[0]; b[1] = bp[192];
    acc = __builtin_amdgcn_wmma_f32_16x16x4_f32(false, a, false, b,
                                                (short)0, acc, false, false);
  }
  float* cp = C + ((size_t)(mtile << 4) + (size_t)(lane >> 4) * 8) * 192 + col;
#pragma unroll
  for (int j = 0; j < 8; ++j) cp[(size_t)j * 192] = acc[j];
}

// ---------------------------------------------------------------------------
// s[n,h] = <xl[n,h,:], att_src[h,:]> ; d[n,h] = <xl[n,h,:], att_dst[h,:]>
// one wave per (n,h); 64 channels -> 2 per lane.
// ---------------------------------------------------------------------------
__global__ void attn_scores(const float* __restrict__ xl,
                            const float* __restrict__ att_src,
                            const float* __restrict__ att_dst,
                            float* __restrict__ s, float* __restrict__ d) {
  const int gw   = (int)((blockIdx.x * blockDim.x + threadIdx.x) >> 5);
  const int lane = threadIdx.x & 31;
  if (gw >= NNODES * 3) return;
  const int n = gw / 3, h = gw % 3;
  const float* xp = xl + (size_t)n * 192 + h * 64;
  const float x0 = xp[lane], x1 = xp[lane + 32];
  float as = att_src[h * 64 + lane] * x0 + att_src[h * 64 + lane + 32] * x1;
  float ad = att_dst[h * 64 + lane] * x0 + att_dst[h * 64 + lane + 32] * x1;
  as = wave_sum32(as);
  ad = wave_sum32(ad);
  if (lane == 0) { s[gw] = as; d[gw] = ad; }
}

// amax = -inf, denom = 0, out accumulator = 0
__global__ void init_accums(float* __restrict__ amax, float* __restrict__ denom,
                            float* __restrict__ outacc, int n_sd, int n_out) {
  const int t = blockIdx.x * blockDim.x + threadIdx.x;
  if (t < n_sd) { amax[t] = -__builtin_inff(); denom[t] = 0.0f; }
  if (t < n_out) outacc[t] = 0.0f;
}

// pass A: alpha = leaky_relu(s[src]+d[dst]); segment-max over dst via
// monotone-bits atomic trick. One thread per (edge, head).
__global__ void edge_alpha_max(const int* __restrict__ src_e,
                               const int* __restrict__ dst_e,
                               const float* __restrict__ s,
                               const float* __restrict__ d,
                               float* __restrict__ alpha,
                               float* __restrict__ amax) {
  const int t = blockIdx.x * blockDim.x + threadIdx.x;
  if (t >= EPLUS * 3) return;
  const int e = t / 3, h = t - e * 3;
  const int srcn = (e < NEDGES) ? src_e[e] : (e - NEDGES);
  const int dstn = (e < NEDGES) ? dst_e[e] : (e - NEDGES);
  float a = s[srcn * 3 + h] + d[dstn * 3 + h];
  a = (a >= 0.0f) ? a : a * NEG_SLOPE;
  alpha[t] = a;
  float* ap = amax + dstn * 3 + h;
  if (a >= 0.0f) atomicMax((int*)ap, __float_as_int(a));
  else           atomicMin((unsigned int*)ap, __float_as_uint(a));
}

// pass B: ex = exp(alpha - amax[dst]); denom[dst] += ex  (ex overwrites alpha)
__global__ void edge_exp_sum(const int* __restrict__ dst_e,
                             const float* __restrict__ amax,
                             float* __restrict__ alpha_ex,
                             float* __restrict__ denom) {
  const int t = blockIdx.x * blockDim.x + threadIdx.x;
  if (t >= EPLUS * 3) return;
  const int e = t / 3, h = t - e * 3;
  const int dstn = (e < NEDGES) ? dst_e[e] : (e - NEDGES);
  const float ex = __expf(alpha_ex[t] - amax[dstn * 3 + h]);
  alpha_ex[t] = ex;
  atomicAdd(denom + dstn * 3 + h, ex);
}

// pass C: out[dst,h,:] += xl[src,h,:] * ex/(denom[dst,h]+1e-16)
// one wave per edge, 6 channels per lane, coalesced loads + atomics (L2-resident).
__global__ void edge_aggregate(const int* __restrict__ src_e,
                               const int* __restrict__ dst_e,
                               const float* __restrict__ xl,
                               const float* __restrict__ ex,
                               const float* __restrict__ denom,
                               float* __restrict__ outacc) {
  const long long gt = (long long)blockIdx.x * blockDim.x + threadIdx.x;
  const int e    = (int)(gt >> 5);
  const int lane = threadIdx.x & 31;
  if (e >= EPLUS) return;
  const int srcn = (e < NEDGES) ? src_e[e] : (e - NEDGES);
  const int dstn = (e < NEDGES) ? dst_e[e] : (e - NEDGES);
  float w[3];
#pragma unroll
  for (int h = 0; h < 3; ++h)
    w[h] = ex[e * 3 + h] / (denom[dstn * 3 + h] + 1e-16f);
  const float* xp = xl + (size_t)srcn * 192;
  float* op = outacc + (size_t)dstn * 192;
#pragma unroll
  for (int i = 0; i < 6; ++i) {
    const int idx = lane + 32 * i;
    atomicAdd(op + idx, xp[idx] * w[idx >> 6]);
  }
}

// layer1 finalize: mean over heads + bias, rrelu (concat=False) -> h1[N,64]
__global__ void finalize1(const float* __restrict__ outacc,
                          const float* __restrict__ bias,
                          float* __restrict__ h1) {
  const int t = blockIdx.x * blockDim.x + threadIdx.x;
  if (t >= NNODES * 64) return;
  const int n = t >> 6, c = t & 63;
  const float* op = outacc + (size_t)n * 192 + c;
  float v = (op[0] + op[64] + op[128]) * (1.0f / 3.0f) + bias[c];
  h1[t] = (v >= 0.0f) ? v : v * RR_SLOPE;
}

// layer2 finalize: +bias, rrelu in-place (concat=True) -> h2[N,192]
__global__ void finalize2(float* __restrict__ outacc,
                          const float* __restrict__ bias) {
  const int t = blockIdx.x * blockDim.x + threadIdx.x;
  if (t >= NNODES * 192) return;
  const float v = outacc[t] + bias[t % 192];
  outacc[t] = (v >= 0.0f) ? v : v * RR_SLOPE;
}

// final linear: out[n] = <h2[n,:], lin_w> + lin_b ; one wave per node
__global__ void final_linear(const float* __restrict__ h2,
                             const float* __restrict__ lin_w,
                             const float* __restrict__ lin_b,
                             float* __restrict__ out) {
  const int gw   = (int)((blockIdx.x * blockDim.x + threadIdx.x) >> 5);
  const int lane = threadIdx.x & 31;
  if (gw >= NNODES) return;
  const float* hp = h2 + (size_t)gw * 192;
  float acc = 0.0f;
#pragma unroll
  for (int i = 0; i < 6; ++i) acc += hp[lane + 32 * i] * lin_w[lane + 32 * i];
  acc = wave_sum32(acc);
  if (lane == 0) out[gw] = acc + lin_b[0];
}

extern "C" void kernel_launch(void* const* d_in, const int* in_sizes, int n_in,
                              void* d_out, int out_size, void* d_ws, size_t ws_size,
                              hipStream_t stream) {
  const float* z        = (const float*)d_in[0];
  const float* x        = (const float*)d_in[1];
  const int*   eidx     = (const int*)d_in[2];
  const float* W1       = (const float*)d_in[3];
  const float* att_src1 = (const float*)d_in[4];
  const float* att_dst1 = (const float*)d_in[5];
  const float* bias1    = (const float*)d_in[6];
  const float* W2       = (const float*)d_in[7];
  const float* att_src2 = (const float*)d_in[8];
  const float* att_dst2 = (const float*)d_in[9];
  const float* bias2    = (const float*)d_in[10];
  const float* lin_w    = (const float*)d_in[11];
  const float* lin_b    = (const float*)d_in[12];
  float* out = (float*)d_out;

  const int* src_e = eidx;
  const int* dst_e = eidx + NEDGES;

  // workspace carve-up (bytes, all 128-aligned by construction)
  char* ws = (char*)d_ws;
  float* XL    = (float*)(ws);                            // N*192 = 76.8 MB
  float* OUT   = (float*)(ws + 76800000);                 // N*192 = 76.8 MB
  float* H1    = (float*)(ws + 153600000);                // N*64  = 25.6 MB
  float* S     = (float*)(ws + 179200000);                // N*3
  float* D     = (float*)(ws + 180400000);                // N*3
  float* AMAX  = (float*)(ws + 181600000);                // N*3
  float* DEN   = (float*)(ws + 182800000);                // N*3
  float* EX    = (float*)(ws + 184000000);                // E'*3 = 20.4 MB

  const int tiles      = (NNODES / 16) * 12;              // 75000 wmma tiles
  const int gemm_blk   = (tiles + 7) / 8;                 // 8 waves / block
  const int sd_blk     = (NNODES * 3 * 32 + 255) / 256;   // wave per (n,h)
  const int init_blk   = (NNODES * 192 + 255) / 256;
  const int eh_blk     = (EPLUS * 3 + 255) / 256;
  const int agg_blk    = (int)(((long long)EPLUS * 32 + 255) / 256);
  const int f1_blk     = (NNODES * 64 + 255) / 256;
  const int f2_blk     = (NNODES * 192 + 255) / 256;
  const int fin_blk    = (NNODES * 32 + 255) / 256;

  // ---------------- layer 1: concat(z,x) -> GAT (mean heads) ----------------
  gemm_wmma_f32<128, true><<<gemm_blk, 256, 0, stream>>>(z, x, W1, XL, NNODES);
  attn_scores<<<sd_blk, 256, 0, stream>>>(XL, att_src1, att_dst1, S, D);
  init_accums<<<init_blk, 256, 0, stream>>>(AMAX, DEN, OUT, NNODES * 3, NNODES * 192);
  edge_alpha_max<<<eh_blk, 256, 0, stream>>>(src_e, dst_e, S, D, EX, AMAX);
  edge_exp_sum<<<eh_blk, 256, 0, stream>>>(dst_e, AMAX, EX, DEN);
  edge_aggregate<<<agg_blk, 256, 0, stream>>>(src_e, dst_e, XL, EX, DEN, OUT);
  finalize1<<<f1_blk, 256, 0, stream>>>(OUT, bias1, H1);

  // ---------------- layer 2: h1 -> GAT (concat heads) ----------------------
  gemm_wmma_f32<64, false><<<gemm_blk, 256, 0, stream>>>(H1, H1, W2, XL, NNODES);
  attn_scores<<<sd_blk, 256, 0, stream>>>(XL, att_src2, att_dst2, S, D);
  init_accums<<<init_blk, 256, 0, stream>>>(AMAX, DEN, OUT, NNODES * 3, NNODES * 192);
  edge_alpha_max<<<eh_blk, 256, 0, stream>>>(src_e, dst_e, S, D, EX, AMAX);
  edge_exp_sum<<<eh_blk, 256, 0, stream>>>(dst_e, AMAX, EX, DEN);
  edge_aggregate<<<agg_blk, 256, 0, stream>>>(src_e, dst_e, XL, EX, DEN, OUT);
  finalize2<<<f2_blk, 256, 0, stream>>>(OUT, bias2);

  // ---------------- final linear -------------------------------------------
  final_linear<<<fin_blk, 256, 0, stream>>>(OUT, lin_w, lin_b, out);
}